// GCNNet_35244501631400
// MI455X (gfx1250) — compile-verified
//
#include <hip/hip_runtime.h>

typedef __attribute__((ext_vector_type(2))) float v2f;
typedef __attribute__((ext_vector_type(8))) float v8f;

#define IN_DIM 116
#define HID    64
#define HC1    32
#define OUTC   2

// ---------------------------------------------------------------- utilities
__global__ void k_fill(float* __restrict__ p, float v, int n) {
  int t = blockIdx.x * blockDim.x + threadIdx.x;
  if (t < n) p[t] = v;
}

// deg[col[e]] += 1  (deg pre-initialized to 1.0 for the self loop)
__global__ void k_deg_edges(const int* __restrict__ col, float* __restrict__ deg, int E) {
  int e = blockIdx.x * blockDim.x + threadIdx.x;
  if (e < E) atomicAdd(&deg[col[e]], 1.0f);
}

// dinv = rsqrt(deg) in place (deg >= 1 always, keep guard for safety)
__global__ void k_dinv(float* __restrict__ deg, int N) {
  int n = blockIdx.x * blockDim.x + threadIdx.x;
  if (n < N) {
    float d = deg[n];
    deg[n] = d > 0.0f ? rsqrtf(d) : 0.0f;
  }
}

// ---------------------------------------------------------------- WMMA GEMM
// C[M x 64] = A[M x K] * B[K x 64], f32 everywhere via V_WMMA_F32_16X16X4_F32.
// One wave per 16x16 tile. blockDim = 128 (4 waves = 4 N-tiles), grid.x = M/16.
// A 16x4 f32 fragment layout (ISA 7.12.2): lanes 0-15 -> K={k0,k0+1},
// lanes 16-31 -> K={k0+2,k0+3}; B 4x16 mirrors the K split across half-waves.
__global__ void k_gemm_x64(const float* __restrict__ A, const float* __restrict__ B,
                           float* __restrict__ C, int M, int K) {
  const int lane  = threadIdx.x & 31;
  const int wave  = threadIdx.x >> 5;
  const int mBase = blockIdx.x << 4;
  const int nBase = wave << 4;
  const int mRow  = mBase + (lane & 15);
  const int kOff  = (lane >> 4) << 1;     // 0 or 2
  const int nCol  = nBase + (lane & 15);

  const float* aRow = A + (size_t)mRow * K;
  v8f acc = {};
  for (int k0 = 0; k0 < K; k0 += 4) {
    v2f a, b;
    a.x = aRow[k0 + kOff];
    a.y = aRow[k0 + kOff + 1];
    b.x = B[(size_t)(k0 + kOff) * HID + nCol];
    b.y = B[(size_t)(k0 + kOff + 1) * HID + nCol];
    acc = __builtin_amdgcn_wmma_f32_16x16x4_f32(
        /*neg_a=*/false, a, /*neg_b=*/false, b,
        /*c_mod=*/(short)0, acc, /*reuse_a=*/false, /*reuse_b=*/false);
  }
  // 16x16 f32 D layout: VGPR v holds M = v (lanes 0-15) / M = v+8 (lanes 16-31)
  const int mHalf = (lane >> 4) << 3;
  float* cBase = C + (size_t)(mBase + mHalf) * HID + nCol;
#pragma unroll
  for (int v = 0; v < 8; ++v) cBase[(size_t)v * HID] = acc[v];
}

// ------------------------------------------------- aggregation (self loop + bias)
// agg[n][c] = dinv[n]^2 * h[n][c] + bias[c]
__global__ void k_agg_init(const float* __restrict__ h, const float* __restrict__ dinv,
                           const float* __restrict__ bias, float* __restrict__ agg, int N) {
  int t = blockIdx.x * blockDim.x + threadIdx.x;
  if (t >= N * HID) return;
  int n = t >> 6, c = t & 63;
  float s = dinv[n];
  agg[t] = s * s * h[t] + bias[c];
}

// per-edge scatter: agg[col] += dinv[row]*dinv[col] * h[row]
// 16 threads per edge, 4 channels (float4) per thread
__global__ void k_scatter(const int* __restrict__ row, const int* __restrict__ col,
                          const float* __restrict__ dinv, const float* __restrict__ h,
                          float* __restrict__ agg, int E) {
  int t  = blockIdx.x * blockDim.x + threadIdx.x;
  int e  = t >> 4;
  int c4 = (t & 15) << 2;
  if (e >= E) return;
  int r = row[e], c = col[e];
  float norm = dinv[r] * dinv[c];
  const float4 hv = *(const float4*)(h + (size_t)r * HID + c4);
  float* dst = agg + (size_t)c * HID + c4;
  atomicAdd(dst + 0, norm * hv.x);
  atomicAdd(dst + 1, norm * hv.y);
  atomicAdd(dst + 2, norm * hv.z);
  atomicAdd(dst + 3, norm * hv.w);
}

__global__ void k_relu(const float* __restrict__ in, float* __restrict__ out, int n) {
  int t = blockIdx.x * blockDim.x + threadIdx.x;
  if (t < n) out[t] = fmaxf(in[t], 0.0f);
}

// ---------------------------------------------------------------- mean pool
__global__ void k_pool(const float* __restrict__ h, const int* __restrict__ batch,
                       float* __restrict__ sums, float* __restrict__ cnt, int N) {
  int t = blockIdx.x * blockDim.x + threadIdx.x;
  if (t >= N * HID) return;
  int n = t >> 6, c = t & 63;
  int g = batch[n];
  atomicAdd(&sums[(size_t)g * HID + c], h[t]);
  if (c == 0) atomicAdd(&cnt[g], 1.0f);
}

// ---------------------------------------------------------------- MLP head
// out[g] = relu(pooled @ Wc1 + bc1) @ Wc2 + bc2 ; one thread per graph
__global__ void k_head(const float* __restrict__ sums, const float* __restrict__ cnt,
                       const float* __restrict__ Wc1, const float* __restrict__ bc1,
                       const float* __restrict__ Wc2, const float* __restrict__ bc2,
                       float* __restrict__ out, int G) {
  int g = blockIdx.x * blockDim.x + threadIdx.x;
  if (g >= G) return;
  float inv = 1.0f / fmaxf(cnt[g], 1.0f);
  float p[HID];
#pragma unroll
  for (int i = 0; i < HID; ++i) p[i] = sums[(size_t)g * HID + i] * inv;
  float o0 = bc2[0], o1 = bc2[1];
  for (int j = 0; j < HC1; ++j) {
    float z = bc1[j];
#pragma unroll
    for (int i = 0; i < HID; ++i) z = fmaf(p[i], Wc1[i * HC1 + j], z);
    z = fmaxf(z, 0.0f);
    o0 = fmaf(z, Wc2[j * OUTC + 0], o0);
    o1 = fmaf(z, Wc2[j * OUTC + 1], o1);
  }
  out[(size_t)g * OUTC + 0] = o0;
  out[(size_t)g * OUTC + 1] = o1;
}

// ---------------------------------------------------------------- launcher
extern "C" void kernel_launch(void* const* d_in, const int* in_sizes, int n_in,
                              void* d_out, int out_size, void* d_ws, size_t ws_size,
                              hipStream_t stream) {
  const float* x    = (const float*)d_in[0];
  const int*   ei   = (const int*)d_in[1];
  const int*   bat  = (const int*)d_in[2];
  const float* W1   = (const float*)d_in[4];
  const float* b1   = (const float*)d_in[5];
  const float* W2   = (const float*)d_in[6];
  const float* b2   = (const float*)d_in[7];
  const float* Wc1  = (const float*)d_in[8];
  const float* bc1  = (const float*)d_in[9];
  const float* Wc2  = (const float*)d_in[10];
  const float* bc2  = (const float*)d_in[11];
  float* out = (float*)d_out;

  const int N = in_sizes[0] / IN_DIM;   // 100000
  const int E = in_sizes[1] / 2;        // 1600000
  const int G = out_size / OUTC;        // 256
  const int* row = ei;                  // edge_index[0] = source
  const int* col = ei + E;              // edge_index[1] = target

  // workspace layout (256B aligned)
  char* ws = (char*)d_ws;
  auto align256 = [](size_t v) { return (v + 255) & ~(size_t)255; };
  const size_t szH = align256((size_t)N * HID * sizeof(float));
  float* bufA = (float*)(ws);                       // 25.6 MB
  float* bufB = (float*)(ws + szH);                 // 25.6 MB
  float* dinv = (float*)(ws + 2 * szH);             // N floats (deg -> dinv)
  float* sums = (float*)(ws + 2 * szH + align256((size_t)N * sizeof(float)));
  float* cnt  = sums + (size_t)G * HID;

  const int NH = N * HID;
  dim3 blk(256);

  // ---- symmetric normalization: deg = 1 (self loop) + count(col); dinv = rsqrt
  k_fill<<<(N + 255) / 256, blk, 0, stream>>>(dinv, 1.0f, N);
  k_deg_edges<<<(E + 255) / 256, blk, 0, stream>>>(col, dinv, E);
  k_dinv<<<(N + 255) / 256, blk, 0, stream>>>(dinv, N);

  // ---- layer 1: h1 = x @ W1 (WMMA f32); agg = dinv^2*h1 + b1 + scatter; relu
  k_gemm_x64<<<N / 16, 128, 0, stream>>>(x, W1, bufA, N, IN_DIM);
  k_agg_init<<<(NH + 255) / 256, blk, 0, stream>>>(bufA, dinv, b1, bufB, N);
  k_scatter<<<(E * 16 + 255) / 256, blk, 0, stream>>>(row, col, dinv, bufA, bufB, E);
  k_relu<<<(NH + 255) / 256, blk, 0, stream>>>(bufB, bufA, NH);

  // ---- layer 2: h2 = r1 @ W2 (WMMA f32); aggregate; relu
  k_gemm_x64<<<N / 16, 128, 0, stream>>>(bufA, W2, bufB, N, HID);
  k_agg_init<<<(NH + 255) / 256, blk, 0, stream>>>(bufB, dinv, b2, bufA, N);
  k_scatter<<<(E * 16 + 255) / 256, blk, 0, stream>>>(row, col, dinv, bufB, bufA, E);
  k_relu<<<(NH + 255) / 256, blk, 0, stream>>>(bufA, bufB, NH);

  // ---- global mean pool + MLP head
  k_fill<<<(G * HID + G + 255) / 256, blk, 0, stream>>>(sums, 0.0f, G * HID + G);
  k_pool<<<(NH + 255) / 256, blk, 0, stream>>>(bufB, bat, sums, cnt, N);
  k_head<<<(G + 255) / 256, blk, 0, stream>>>(sums, cnt, Wc1, bc1, Wc2, bc2, out, G);
}